// GlobalGCN_36172214567701
// MI455X (gfx1250) — compile-verified
//
#include <hip/hip_runtime.h>
#include <hip/hip_bf16.h>

typedef float v2f __attribute__((ext_vector_type(2)));
typedef float v8f __attribute__((ext_vector_type(8)));

#define DDIM 256

// ---------------------------------------------------------------------------
// Kernel 1: support = X @ W^T using V_WMMA_F32_16X16X4_F32 (fp32 exact).
// Each wave computes a 16(M) x 64(N) strip: 4 accumulators, A-frag reused 4x.
// A(16x4 f32): lane<16 -> (M=lane, K=vgpr), lane>=16 -> (M=lane-16, K=2+vgpr)
// B(4x16 f32): same pattern with N instead of M; since B = W^T and W is
// row-major (out,in), both fragments load 2 contiguous floats per lane.
// ---------------------------------------------------------------------------
__global__ __launch_bounds__(256) void gemm_wmma_f32(const float* __restrict__ X,
                                                     const float* __restrict__ W,
                                                     float* __restrict__ S,
                                                     int M, int mtiles) {
    const int wave = threadIdx.x >> 5;
    const int lane = threadIdx.x & 31;
    const int half = lane >> 4;          // 0/1: selects K{0,1} vs K{2,3}
    const int r    = lane & 15;

    int t = blockIdx.x * 8 + wave;       // wave-tile id
    int n_group = t & 3;                 // 4 N-groups of 64 cols
    int m_tile  = t >> 2;
    if (m_tile >= mtiles) return;        // uniform per wave -> EXEC stays full

    const int row0 = m_tile * 16;
    const int col0 = n_group * 64;

    int ar = row0 + r; if (ar > M - 1) ar = M - 1;   // clamp (M%16==0 normally)

    const float* aptr = X + (size_t)ar * DDIM + 2 * half;
    const float* bptr = W + (size_t)(col0 + r) * DDIM + 2 * half;

    v8f acc0 = {0,0,0,0,0,0,0,0};
    v8f acc1 = {0,0,0,0,0,0,0,0};
    v8f acc2 = {0,0,0,0,0,0,0,0};
    v8f acc3 = {0,0,0,0,0,0,0,0};

#pragma unroll 4
    for (int k = 0; k < DDIM; k += 4) {
        v2f a  = *(const v2f*)(aptr + k);
        v2f b0 = *(const v2f*)(bptr + k);
        v2f b1 = *(const v2f*)(bptr + 16 * DDIM + k);
        v2f b2 = *(const v2f*)(bptr + 32 * DDIM + k);
        v2f b3 = *(const v2f*)(bptr + 48 * DDIM + k);
        acc0 = __builtin_amdgcn_wmma_f32_16x16x4_f32(false, a, false, b0, (short)0, acc0, false, false);
        acc1 = __builtin_amdgcn_wmma_f32_16x16x4_f32(false, a, false, b1, (short)0, acc1, false, false);
        acc2 = __builtin_amdgcn_wmma_f32_16x16x4_f32(false, a, false, b2, (short)0, acc2, false, false);
        acc3 = __builtin_amdgcn_wmma_f32_16x16x4_f32(false, a, false, b3, (short)0, acc3, false, false);
    }

    // C/D layout: VGPR v: lane<16 -> M=v, lane>=16 -> M=v+8; N = lane&15
    float* cbase = S + (size_t)row0 * DDIM + col0;
#pragma unroll
    for (int v = 0; v < 8; v++) {
        int m = v + 8 * half;
        if (row0 + m < M) {
            cbase[(size_t)m * DDIM +  0 + r] = acc0[v];
            cbase[(size_t)m * DDIM + 16 + r] = acc1[v];
            cbase[(size_t)m * DDIM + 32 + r] = acc2[v];
            cbase[(size_t)m * DDIM + 48 + r] = acc3[v];
        }
    }
}

// ---------------------------------------------------------------------------
// CSR construction: histogram -> exclusive scan -> scatter -> per-row sort.
// ---------------------------------------------------------------------------
__global__ void hist_kernel(const int* __restrict__ erow, int* __restrict__ cnt, int E) {
    int e = blockIdx.x * blockDim.x + threadIdx.x;
    if (e < E) atomicAdd(&cnt[erow[e]], 1);
}

__global__ void scan_kernel(const int* __restrict__ cnt, int* __restrict__ rowptr, int Nn) {
    __shared__ int sums[1024];
    __shared__ int offs[1024];
    int tid = threadIdx.x;
    int chunk = (Nn + 1023) >> 10;
    int start = tid * chunk;
    int end = start + chunk; if (end > Nn) end = Nn;
    int s = 0;
    if (start < Nn) for (int i = start; i < end; i++) s += cnt[i];
    sums[tid] = s;
    __syncthreads();
    if (tid == 0) {
        int run = 0;
        for (int i = 0; i < 1024; i++) { offs[i] = run; run += sums[i]; }
    }
    __syncthreads();
    if (start < Nn) {
        int run = offs[tid];
        for (int i = start; i < end; i++) { rowptr[i] = run; run += cnt[i]; }
        if (end == Nn) rowptr[Nn] = run;   // == E
    }
}

__global__ void scatter_kernel(const int* __restrict__ erow, const int* __restrict__ rowptr,
                               int* __restrict__ cur, int* __restrict__ perm, int E) {
    int e = blockIdx.x * blockDim.x + threadIdx.x;
    if (e >= E) return;
    int r = erow[e];
    int p = rowptr[r] + atomicAdd(&cur[r], 1);
    perm[p] = e;
}

// Canonicalize bucket order (ascending edge id) -> deterministic fp sums.
__global__ void rowsort_kernel(const int* __restrict__ rowptr, int* __restrict__ perm, int Nn) {
    int row = blockIdx.x * blockDim.x + threadIdx.x;
    if (row >= Nn) return;
    int s = rowptr[row], e = rowptr[row + 1];
    for (int i = s + 1; i < e; i++) {
        int key = perm[i];
        int j = i - 1;
        while (j >= s && perm[j] > key) { perm[j + 1] = perm[j]; j--; }
        perm[j + 1] = key;
    }
}

// ---------------------------------------------------------------------------
// CSR SpMM: out[row] = sum_e val[e] * support[col[e]].  64 threads per row,
// float4 per thread. Gathered reads hit L2 (support = 102 MB < 192 MB L2),
// writes fully coalesced, zero float atomics, deterministic order.
// ---------------------------------------------------------------------------
__global__ __launch_bounds__(256) void spmm_csr(const float* __restrict__ S,
                                                const int* __restrict__ rowptr,
                                                const int* __restrict__ perm,
                                                const int* __restrict__ ecol,
                                                const float* __restrict__ eval,
                                                float* __restrict__ out, int Nn) {
    int row = blockIdx.x * 4 + (threadIdx.x >> 6);
    if (row >= Nn) return;
    int c = (threadIdx.x & 63) * 4;
    float4 acc = make_float4(0.f, 0.f, 0.f, 0.f);
    int s = rowptr[row], e = rowptr[row + 1];
    for (int i = s; i < e; i++) {
        int eid = perm[i];
        float v = eval[eid];
        const float4 sv = *(const float4*)(S + (size_t)ecol[eid] * DDIM + c);
        acc.x += v * sv.x;
        acc.y += v * sv.y;
        acc.z += v * sv.z;
        acc.w += v * sv.w;
    }
    *(float4*)(out + (size_t)row * DDIM + c) = acc;
}

extern "C" void kernel_launch(void* const* d_in, const int* in_sizes, int n_in,
                              void* d_out, int out_size, void* d_ws, size_t ws_size,
                              hipStream_t stream) {
    const float* x    = (const float*)d_in[0];
    const float* W    = (const float*)d_in[1];
    const int*   erow = (const int*)d_in[2];
    const int*   ecol = (const int*)d_in[3];
    const float* eval = (const float*)d_in[4];

    const int M = in_sizes[0] / DDIM;   // number of nodes
    const int E = in_sizes[2];          // number of edges

    char* ws = (char*)d_ws;
    size_t off = 0;
    float* S      = (float*)(ws + off); off += (size_t)M * DDIM * sizeof(float);
    int*   rowptr = (int*)(ws + off);   off += (size_t)(M + 1) * sizeof(int);
    int*   cnt    = (int*)(ws + off);   off += (size_t)M * sizeof(int);
    int*   cur    = (int*)(ws + off);   off += (size_t)M * sizeof(int);
    int*   perm   = (int*)(ws + off);   off += (size_t)E * sizeof(int);

    hipMemsetAsync(d_out, 0, (size_t)out_size * sizeof(float), stream);
    hipMemsetAsync(cnt, 0, (size_t)M * sizeof(int), stream);
    hipMemsetAsync(cur, 0, (size_t)M * sizeof(int), stream);

    // GEMM: wave = 16x64 strip; 8 waves per block.
    int mtiles = (M + 15) / 16;
    int total_waves = mtiles * 4;
    int gblocks = (total_waves + 7) / 8;
    gemm_wmma_f32<<<gblocks, 256, 0, stream>>>(x, W, S, M, mtiles);

    hist_kernel<<<(E + 255) / 256, 256, 0, stream>>>(erow, cnt, E);
    scan_kernel<<<1, 1024, 0, stream>>>(cnt, rowptr, M);
    scatter_kernel<<<(E + 255) / 256, 256, 0, stream>>>(erow, rowptr, cur, perm, E);
    rowsort_kernel<<<(M + 255) / 256, 256, 0, stream>>>(rowptr, perm, M);
    spmm_csr<<<(M + 3) / 4, 256, 0, stream>>>(S, rowptr, perm, ecol, eval,
                                              (float*)d_out, M);
}